// BiClassifier_27625229648164
// MI455X (gfx1250) — compile-verified
//
#include <hip/hip_runtime.h>
#include <hip/hip_bf16.h>

typedef __attribute__((ext_vector_type(2))) float v2f;
typedef __attribute__((ext_vector_type(8))) float v8f;

#define HID   1024
#define DIN   768
#define BATCH 4
#define NSEQ  128

// ---------------------------------------------------------------------------
// Phase 1: Y[M,N] = X[M,K] @ W[N,K]^T (+ bias[N]), fp32 WMMA 16x16x4.
// One wave per 16x64 output strip: one A fragment feeds 4 WMMAs per k-step
// (4 independent accumulators -> no D->A/B hazard chain, 10 B/lane/wmma).
// Compile-time dims so addressing folds to immediates (cuts SALU).
// ---------------------------------------------------------------------------
__global__ __launch_bounds__(32) void bicls_gemm_wmma_f32(
    const float* __restrict__ X,     // [M, DIN]
    const float* __restrict__ W,     // [HID, DIN]  (torch Linear: [out, in])
    const float* __restrict__ bias,  // [HID] or nullptr
    float* __restrict__ Y)           // [M, HID]
{
    constexpr int K = DIN;
    constexpr int N = HID;

    const int lane = threadIdx.x;      // 0..31, EXEC all-ones (exact grid)
    const int half = lane >> 4;        // which K pair within the group of 4
    const int l16  = lane & 15;

    const int h0 = blockIdx.x * 64;    // N strip: 4 consecutive 16-wide tiles
    const int m0 = blockIdx.y * 16;    // M tile

    // A: lane holds X[m0+l16, k + half*2 + {0,1}]     (contiguous float2)
    // B(t): lane holds W[h0+16t+l16, k + half*2 + {0,1}]
    const float* xr = X + (size_t)(m0 + l16) * K + half * 2;
    const float* w0 = W + (size_t)(h0 +  0 + l16) * K + half * 2;
    const float* w1 = W + (size_t)(h0 + 16 + l16) * K + half * 2;
    const float* w2 = W + (size_t)(h0 + 32 + l16) * K + half * 2;
    const float* w3 = W + (size_t)(h0 + 48 + l16) * K + half * 2;

    // C/D layout: c[j] = C[m0 + j + half*8][h + l16]; bias is per-column ->
    // one scalar per lane per tile, replicated across the 8 accumulator regs.
    const float bv0 = bias ? bias[h0 +  0 + l16] : 0.0f;
    const float bv1 = bias ? bias[h0 + 16 + l16] : 0.0f;
    const float bv2 = bias ? bias[h0 + 32 + l16] : 0.0f;
    const float bv3 = bias ? bias[h0 + 48 + l16] : 0.0f;
    v8f c0, c1, c2, c3;
    #pragma unroll
    for (int j = 0; j < 8; ++j) { c0[j] = bv0; c1[j] = bv1; c2[j] = bv2; c3[j] = bv3; }

    #pragma unroll 4
    for (int k = 0; k < K; k += 4) {
        const v2f a  = *(const v2f*)(xr + k);
        const v2f b0 = *(const v2f*)(w0 + k);
        const v2f b1 = *(const v2f*)(w1 + k);
        const v2f b2 = *(const v2f*)(w2 + k);
        const v2f b3 = *(const v2f*)(w3 + k);
        // 8 args: (neg_a, A, neg_b, B, c_mod, C, reuse_a, reuse_b)
        c0 = __builtin_amdgcn_wmma_f32_16x16x4_f32(false, a, false, b0, (short)0, c0, false, false);
        c1 = __builtin_amdgcn_wmma_f32_16x16x4_f32(false, a, false, b1, (short)0, c1, false, false);
        c2 = __builtin_amdgcn_wmma_f32_16x16x4_f32(false, a, false, b2, (short)0, c2, false, false);
        c3 = __builtin_amdgcn_wmma_f32_16x16x4_f32(false, a, false, b3, (short)0, c3, false, false);
    }

    // Store: per j, each half-wave writes a 64B coalesced segment.
    float* yout = Y + (size_t)m0 * N + l16;
    #pragma unroll
    for (int j = 0; j < 8; ++j) {
        float* row = yout + (size_t)(j + half * 8) * N;
        row[h0 +  0] = c0[j];
        row[h0 + 16] = c1[j];
        row[h0 + 32] = c2[j];
        row[h0 + 48] = c3[j];
    }
}

// ---------------------------------------------------------------------------
// Phase 2: out[b,n,m,o] = sum_h relu(in1[b,n,h] + in2[b,m,h]) * Wo[o,h] + bo[o]
// 16x16 pair tile per block; h streamed in 128-chunks through padded LDS.
// OUT=2 would waste 14/16 of a WMMA tile's N dim and the relu-formation is
// inherently elementwise, so this stays on the VALU with fma-dense loops.
// ---------------------------------------------------------------------------
#define HC 128
#define PAD 4   // row pitch 132 floats = 528 B: lane stride 16 B -> distinct banks

__global__ __launch_bounds__(256) void bicls_pairwise(
    const float* __restrict__ A1,   // [B*N1, HID]  (in1, bias applied)
    const float* __restrict__ A2,   // [B*N2, HID]  (in2)
    const float* __restrict__ Wo,   // [2, HID]
    const float* __restrict__ bo,   // [2]
    float* __restrict__ out)        // [B, N1, N2, 2]
{
    __shared__ __align__(16) float s1[16][HC + PAD];
    __shared__ __align__(16) float s2[16][HC + PAD];
    __shared__ __align__(16) float sw0[HC];
    __shared__ __align__(16) float sw1[HC];

    const int b  = blockIdx.z;
    const int n0 = blockIdx.y * 16;
    const int m0 = blockIdx.x * 16;

    const int tid = threadIdx.x;   // 0..255
    const int tx  = tid & 15;      // m within tile
    const int ty  = tid >> 4;      // n within tile

    const float* arow = A1 + ((size_t)b * NSEQ + n0) * HID;
    const float* crow = A2 + ((size_t)b * NSEQ + m0) * HID;

    float acc0 = 0.0f, acc1 = 0.0f;

    for (int h0 = 0; h0 < HID; h0 += HC) {
        // Cooperative fill: 16 rows x 128 cols per buffer; 8 contiguous
        // floats per thread per buffer.
        {
            const int r  = tid >> 4;
            const int c0 = (tid & 15) * (HC / 16);
            #pragma unroll
            for (int i = 0; i < HC / 16; ++i) {
                s1[r][c0 + i] = arow[(size_t)r * HID + h0 + c0 + i];
                s2[r][c0 + i] = crow[(size_t)r * HID + h0 + c0 + i];
            }
            if (tid < HC) {
                sw0[tid] = Wo[h0 + tid];
                sw1[tid] = Wo[HID + h0 + tid];
            }
        }
        __syncthreads();

        #pragma unroll 8
        for (int hh = 0; hh < HC; hh += 4) {
            const float4 x1 = *(const float4*)(&s1[ty][hh]);
            const float4 x2 = *(const float4*)(&s2[tx][hh]);
            const float4 w0 = *(const float4*)(&sw0[hh]);
            const float4 w1 = *(const float4*)(&sw1[hh]);
            float v;
            v = fmaxf(x1.x + x2.x, 0.0f); acc0 = fmaf(v, w0.x, acc0); acc1 = fmaf(v, w1.x, acc1);
            v = fmaxf(x1.y + x2.y, 0.0f); acc0 = fmaf(v, w0.y, acc0); acc1 = fmaf(v, w1.y, acc1);
            v = fmaxf(x1.z + x2.z, 0.0f); acc0 = fmaf(v, w0.z, acc0); acc1 = fmaf(v, w1.z, acc1);
            v = fmaxf(x1.w + x2.w, 0.0f); acc0 = fmaf(v, w0.w, acc0); acc1 = fmaf(v, w1.w, acc1);
        }
        __syncthreads();
    }

    const int n = n0 + ty;
    const int m = m0 + tx;
    float2 r;
    r.x = acc0 + bo[0];
    r.y = acc1 + bo[1];
    *(float2*)(out + (((size_t)b * NSEQ + n) * NSEQ + m) * 2) = r;
}

// ---------------------------------------------------------------------------
extern "C" void kernel_launch(void* const* d_in, const int* in_sizes, int n_in,
                              void* d_out, int out_size, void* d_ws, size_t ws_size,
                              hipStream_t stream) {
    const float* input1 = (const float*)d_in[0];  // [4,128,768]
    const float* input2 = (const float*)d_in[1];  // [4,128,768]
    const float* W1     = (const float*)d_in[2];  // [1024,768]
    const float* b1     = (const float*)d_in[3];  // [1024]
    const float* W2     = (const float*)d_in[4];  // [1024,768]
    const float* Wo     = (const float*)d_in[5];  // [2,1024]
    const float* bo     = (const float*)d_in[6];  // [2]
    float* out = (float*)d_out;

    const int M = BATCH * NSEQ;                   // 512
    float* ws1 = (float*)d_ws;                    // [512,1024]
    float* ws2 = ws1 + (size_t)M * HID;           // [512,1024] (4 MB total)

    dim3 gGemm(HID / 64, M / 16);                 // (16, 32) 16x64-strip waves
    bicls_gemm_wmma_f32<<<gGemm, 32, 0, stream>>>(input1, W1, b1,      ws1);
    bicls_gemm_wmma_f32<<<gGemm, 32, 0, stream>>>(input2, W2, nullptr, ws2);

    dim3 gPair(NSEQ / 16, NSEQ / 16, BATCH);      // (8, 8, 4)
    bicls_pairwise<<<gPair, 256, 0, stream>>>(ws1, ws2, Wo, bo, out);
}